// SRVSKG_11355893530827
// MI455X (gfx1250) — compile-verified
//
#include <hip/hip_runtime.h>
#include <math.h>

#define N_ENT 50000
#define D     128
#define E_DIM 32
#define N_VIRT 3
#define E_ATT 600000
#define E_KG  600000

typedef __attribute__((ext_vector_type(2))) float v2f;
typedef __attribute__((ext_vector_type(8))) float v8f;

__device__ __forceinline__ float wave_sum32(float v) {
#pragma unroll
  for (int o = 16; o > 0; o >>= 1) v += __shfl_xor(v, o);
  return v;
}

// ---------------- 1. LayerNorm (no affine), wave per row -------------------
__global__ void srv_ln(const float* __restrict__ x, float* __restrict__ yn) {
  int row  = (blockIdx.x * blockDim.x + threadIdx.x) >> 5;
  int lane = threadIdx.x & 31;
  const float4 v = reinterpret_cast<const float4*>(x + (size_t)row * D)[lane];
  float mu  = wave_sum32(v.x + v.y + v.z + v.w) * (1.0f / D);
  float dx = v.x - mu, dy = v.y - mu, dz = v.z - mu, dw = v.w - mu;
  float var = wave_sum32(dx*dx + dy*dy + dz*dz + dw*dw) * (1.0f / D);
  float inv = rsqrtf(var + 1e-5f);
  float4 o; o.x = dx*inv; o.y = dy*inv; o.z = dz*inv; o.w = dw*inv;
  reinterpret_cast<float4*>(yn + (size_t)row * D)[lane] = o;
}

// ---------------- 2. Attention scores: 16 edges / wave via WMMA ------------
// D[e][e] of (Q_batch × K_batch^T) gives the 16 per-edge 128-d dots.
// A: 16x4 f32 (lane m<16 holds M=m,K={0,1}; lane m+16 holds M=m,K={2,3})
// B: 4x16 f32 (mirrored K split across lane halves), C/D: 16x16 f32, 8 VGPRs.
__global__ void srv_att_score(const float* __restrict__ yn,
                              const float* __restrict__ eigs,
                              const float* __restrict__ lam,
                              const int*   __restrict__ att,
                              float* __restrict__ vals,
                              float* __restrict__ denom) {
  __shared__ float qs[2][16][129];   // bank-pad stride 129
  __shared__ float ks[2][16][129];
  const int wslot = threadIdx.x >> 5;
  const int lane  = threadIdx.x & 31;
  const int sub   = lane & 15;
  const int half  = lane >> 4;
  const int e0    = (blockIdx.x * 2 + wslot) * 16;   // exact grid: no tail
  const int eidx  = e0 + sub;
  const int src   = att[eidx];
  const int dst   = att[E_ATT + eidx];

  // Stage the 16 q-rows and 16 k-rows of this wave's edges into LDS (coalesced).
#pragma unroll 4
  for (int r = 0; r < 16; ++r) {
    int sr = __shfl(src, r);
    int dr = __shfl(dst, r);
    float4 qv = reinterpret_cast<const float4*>(yn + (size_t)sr * D)[lane];
    float4 kv = reinterpret_cast<const float4*>(yn + (size_t)dr * D)[lane];
    qs[wslot][r][lane*4+0] = qv.x; qs[wslot][r][lane*4+1] = qv.y;
    qs[wslot][r][lane*4+2] = qv.z; qs[wslot][r][lane*4+3] = qv.w;
    ks[wslot][r][lane*4+0] = kv.x; ks[wslot][r][lane*4+1] = kv.y;
    ks[wslot][r][lane*4+2] = kv.z; ks[wslot][r][lane*4+3] = kv.w;
  }
  __syncthreads();

  v8f acc = {0.f,0.f,0.f,0.f,0.f,0.f,0.f,0.f};
  const int kofs = 2 * half;
#pragma unroll 4
  for (int c = 0; c < D; c += 4) {
    v2f a, b;
    a.x = qs[wslot][sub][c + kofs];     // A[M=sub][K=kofs]
    a.y = qs[wslot][sub][c + kofs + 1];
    b.x = ks[wslot][sub][c + kofs];     // B[K=kofs][N=sub]
    b.y = ks[wslot][sub][c + kofs + 1];
    acc = __builtin_amdgcn_wmma_f32_16x16x4_f32(false, a, false, b,
                                                (short)0, acc, false, false);
  }

  // Extract diagonal D[e][e] -> lane (e mod 16) of each half.
  float x = 0.f;
#pragma unroll
  for (int r = 0; r < 8; ++r) {
    float dlo = __shfl((float)acc[r], r);        // D[r][r]
    float dhi = __shfl((float)acc[r], 24 + r);   // D[8+r][8+r]
    if (sub == r)     x = dlo;
    if (sub == r + 8) x = dhi;
  }

  // 32-d eig similarity: halves split the 32 dims, then xor-combine.
  float y = 0.f;
  const float* es = eigs + (size_t)src * E_DIM + half * 16;
  const float* ed = eigs + (size_t)dst * E_DIM + half * 16;
#pragma unroll
  for (int j = 0; j < 16; ++j) y += es[j] * ed[j];
  y += __shfl_xor(y, 16);

  float s   = x * 0.08838834764831845f /* 1/sqrt(128) */ + expf(lam[0]) * y;
  float val = fminf(fmaxf(expf(s), -5.0f), 5.0f);
  if (half == 0) {
    vals[eidx] = val;
    atomicAdd(&denom[src], val);
  }
}

// ---------------- 3. Attention aggregation: wave per edge ------------------
__global__ void srv_att_agg(const float* __restrict__ yn,
                            const int*   __restrict__ att,
                            const float* __restrict__ vals,
                            const float* __restrict__ denom,
                            float* __restrict__ enc) {
  int e    = (blockIdx.x * blockDim.x + threadIdx.x) >> 5;
  int lane = threadIdx.x & 31;
  int src = att[e], dst = att[E_ATT + e];
  float dn = denom[src];
  float sn = vals[e] / (dn == 0.0f ? 1.0f : dn);
  float4 v = reinterpret_cast<const float4*>(yn + (size_t)dst * D)[lane];
  float* o = enc + (size_t)src * D + lane * 4;
  atomicAdd(o + 0, sn * v.x); atomicAdd(o + 1, sn * v.y);
  atomicAdd(o + 2, sn * v.z); atomicAdd(o + 3, sn * v.w);
}

// ---------------- 4. Routing helpers --------------------------------------
__global__ void srv_cnt(const int* __restrict__ kg, const int* __restrict__ kt,
                        int t, float* __restrict__ cnt) {
  int e = blockIdx.x * blockDim.x + threadIdx.x;
  if (e >= E_KG) return;
  if (kt[e] == t) atomicAdd(&cnt[kg[e]], 1.0f);
}

__global__ void srv_esc_init(float* __restrict__ esc) {
  int e = blockIdx.x * blockDim.x + threadIdx.x;
  if (e < E_KG) esc[e] = 1.0f;
}

// sim = dot(u[head], tanh(esc*enc[tail])); esc *= sim. Wave per edge.
__global__ void srv_sim(const float* __restrict__ u, const float* __restrict__ enc,
                        const int* __restrict__ kg, const int* __restrict__ kt,
                        int t, float* __restrict__ esc) {
  int e    = (blockIdx.x * blockDim.x + threadIdx.x) >> 5;
  int lane = threadIdx.x & 31;
  if (kt[e] != t) return;                      // uniform per wave
  int head = kg[e], tail = kg[E_KG + e];
  float sc = esc[e];
  float4 uv = reinterpret_cast<const float4*>(u   + (size_t)head * D)[lane];
  float4 nv = reinterpret_cast<const float4*>(enc + (size_t)tail * D)[lane];
  float p = uv.x * tanhf(sc * nv.x) + uv.y * tanhf(sc * nv.y)
          + uv.z * tanhf(sc * nv.z) + uv.w * tanhf(sc * nv.w);
  p = wave_sum32(p);
  if (lane == 0) esc[e] = sc * p;
}

// agg[head] += esc[e] * enc[tail] for edges of this type. Wave per edge.
__global__ void srv_scatter(const float* __restrict__ enc,
                            const int* __restrict__ kg, const int* __restrict__ kt,
                            int t, const float* __restrict__ esc,
                            float* __restrict__ agg) {
  int e    = (blockIdx.x * blockDim.x + threadIdx.x) >> 5;
  int lane = threadIdx.x & 31;
  if (kt[e] != t) return;
  int head = kg[e], tail = kg[E_KG + e];
  float sc = esc[e];
  float4 nv = reinterpret_cast<const float4*>(enc + (size_t)tail * D)[lane];
  float* o = agg + (size_t)head * D + lane * 4;
  atomicAdd(o + 0, sc * nv.x); atomicAdd(o + 1, sc * nv.y);
  atomicAdd(o + 2, sc * nv.z); atomicAdd(o + 3, sc * nv.w);
}

// u = squash(agg/max(cnt,1)) + enc   (wave per row)
__global__ void srv_combine(const float* __restrict__ agg, const float* __restrict__ cnt,
                            const float* __restrict__ enc, float* __restrict__ u) {
  int n    = (blockIdx.x * blockDim.x + threadIdx.x) >> 5;
  int lane = threadIdx.x & 31;
  float c = fmaxf(cnt[n], 1.0f);
  float4 a = reinterpret_cast<const float4*>(agg + (size_t)n * D)[lane];
  a.x /= c; a.y /= c; a.z /= c; a.w /= c;
  float ss  = wave_sum32(a.x*a.x + a.y*a.y + a.z*a.z + a.w*a.w);
  float nrm = sqrtf(ss);
  float f   = (nrm * nrm / (nrm * nrm + 1.0f)) / fmaxf(nrm, 1e-12f);
  float4 e = reinterpret_cast<const float4*>(enc + (size_t)n * D)[lane];
  float4 o; o.x = f*a.x + e.x; o.y = f*a.y + e.y; o.z = f*a.z + e.z; o.w = f*a.w + e.w;
  reinterpret_cast<float4*>(u + (size_t)n * D)[lane] = o;
}

// out += softmax(w)[t] * (agg/max(cnt,1) + enc)   (last iter: no squash)
__global__ void srv_final(const float* __restrict__ agg, const float* __restrict__ cnt,
                          const float* __restrict__ enc, const float* __restrict__ w,
                          int t, float* __restrict__ out) {
  int n    = (blockIdx.x * blockDim.x + threadIdx.x) >> 5;
  int lane = threadIdx.x & 31;
  float w0 = w[0], w1 = w[1], w2 = w[2];
  float m  = fmaxf(w0, fmaxf(w1, w2));
  float e0 = expf(w0 - m), e1 = expf(w1 - m), e2 = expf(w2 - m);
  float wt = (t == 0 ? e0 : t == 1 ? e1 : e2) / (e0 + e1 + e2);
  float c = fmaxf(cnt[n], 1.0f);
  float4 a = reinterpret_cast<const float4*>(agg + (size_t)n * D)[lane];
  float4 e = reinterpret_cast<const float4*>(enc + (size_t)n * D)[lane];
  float* o = out + (size_t)n * D + lane * 4;
  o[0] += wt * (a.x / c + e.x); o[1] += wt * (a.y / c + e.y);
  o[2] += wt * (a.z / c + e.z); o[3] += wt * (a.w / c + e.w);
}

// ---------------------------------------------------------------------------
extern "C" void kernel_launch(void* const* d_in, const int* in_sizes, int n_in,
                              void* d_out, int out_size, void* d_ws, size_t ws_size,
                              hipStream_t stream) {
  const float* emb   = (const float*)d_in[0];
  const float* eigs  = (const float*)d_in[1];
  const float* lam   = (const float*)d_in[2];
  const float* w     = (const float*)d_in[3];
  const int*   att   = (const int*)d_in[4];
  const int*   kg    = (const int*)d_in[5];
  const int*   ktype = (const int*)d_in[6];
  float* out = (float*)d_out;

  const size_t ND = (size_t)N_ENT * D;
  float* yn    = (float*)d_ws;
  float* enc   = yn    + ND;
  float* vals  = enc   + ND;
  float* denom = vals  + E_ATT;
  float* agg   = denom + N_ENT;
  float* cnt   = agg   + ND;
  float* esc   = cnt   + N_ENT;
  float* u     = esc   + E_KG;

  // Encoder: layernorm -> sparse attention
  srv_ln<<<N_ENT / 8, 256, 0, stream>>>(emb, yn);                       // 6250 blocks
  hipMemsetAsync(denom, 0, N_ENT * sizeof(float), stream);
  srv_att_score<<<E_ATT / 32, 64, 0, stream>>>(yn, eigs, lam, att, vals, denom); // 18750
  hipMemsetAsync(enc, 0, ND * sizeof(float), stream);
  srv_att_agg<<<E_ATT / 8, 256, 0, stream>>>(yn, att, vals, denom, enc); // 75000

  // Aggregator: per-type capsule routing, softmax(w)-weighted sum into out
  hipMemsetAsync(out, 0, (size_t)out_size * sizeof(float), stream);
  for (int t = 0; t < N_VIRT; ++t) {
    hipMemsetAsync(cnt, 0, N_ENT * sizeof(float), stream);
    srv_cnt<<<(E_KG + 255) / 256, 256, 0, stream>>>(kg, ktype, t, cnt);
    srv_esc_init<<<(E_KG + 255) / 256, 256, 0, stream>>>(esc);
    // iter 0
    hipMemsetAsync(agg, 0, ND * sizeof(float), stream);
    srv_scatter<<<E_KG / 8, 256, 0, stream>>>(enc, kg, ktype, t, esc, agg);
    srv_combine<<<N_ENT / 8, 256, 0, stream>>>(agg, cnt, enc, u);
    // iter 1
    srv_sim<<<E_KG / 8, 256, 0, stream>>>(u, enc, kg, ktype, t, esc);
    hipMemsetAsync(agg, 0, ND * sizeof(float), stream);
    srv_scatter<<<E_KG / 8, 256, 0, stream>>>(enc, kg, ktype, t, esc, agg);
    srv_combine<<<N_ENT / 8, 256, 0, stream>>>(agg, cnt, enc, u);
    // iter 2 (no squash; accumulate weighted into out)
    srv_sim<<<E_KG / 8, 256, 0, stream>>>(u, enc, kg, ktype, t, esc);
    hipMemsetAsync(agg, 0, ND * sizeof(float), stream);
    srv_scatter<<<E_KG / 8, 256, 0, stream>>>(enc, kg, ktype, t, esc, agg);
    srv_final<<<N_ENT / 8, 256, 0, stream>>>(agg, cnt, enc, w, t, out);
  }
}